// HybridNatEstimatorSampler_65481071403039
// MI455X (gfx1250) — compile-verified
//
#include <hip/hip_runtime.h>

typedef float f32x2 __attribute__((ext_vector_type(2)));
typedef float f32x8 __attribute__((ext_vector_type(8)));

// ---------------------------------------------------------------------------
// Kernel 1: fused conv1 -> relu -> maxpool2 -> conv2(WMMA f32 16x16x4)
//           -> +bias -> relu -> maxpool2 -> 6x6 mean  => pooled[B,16]
// One wave32 per batch element; 2 waves (64 threads) per block.
// ---------------------------------------------------------------------------
namespace {
constexpr int WPB  = 2;          // waves per block
constexpr int XSSZ = 30 * 32;    // x staged with 1-halo: [30][32]
constexpr int H1SZ = 8 * 15 * 16;// conv1 output with 1-halo: [8][15][16]
constexpr int C2SZ = 144 * 16;   // conv2 output: [144 px][16 ch]
constexpr int PERW = XSSZ + H1SZ + C2SZ; // 5184 floats per wave
}

__global__ __launch_bounds__(64)
void fused_cnn_kernel(const float* __restrict__ x,
                      const float* __restrict__ w1g, const float* __restrict__ b1g,
                      const float* __restrict__ w2g, const float* __restrict__ b2g,
                      float* __restrict__ pooled, int B)
{
    __shared__ float wsh[96];              // 72 conv1_w + 8 conv1_b + 16 conv2_b
    __shared__ float lds[WPB * PERW];

    const int tid  = threadIdx.x;
    const int wave = tid >> 5;
    const int lane = tid & 31;
    const int b    = blockIdx.x * WPB + wave;

    // prefetch next block's image data (global_prefetch_b8)
    if (tid == 0 && (blockIdx.x + 1) * WPB < B)
        __builtin_prefetch(x + (size_t)(blockIdx.x + 1) * WPB * 784, 0, 1);

    // stage the tiny weight vectors once per block
    for (int i = tid; i < 96; i += 64) {
        float v;
        if (i < 72)      v = w1g[i];
        else if (i < 80) v = b1g[i - 72];
        else             v = b2g[i - 80];
        wsh[i] = v;
    }
    __syncthreads();

    float* xs  = lds + wave * PERW;   // [30][32], zero halo
    float* h1p = xs + XSSZ;           // [8][15][16], zero halo at row0/col0
    float* c2s = h1p + H1SZ;          // [144][16]

    // ---- stage x with halo ----
    for (int i = lane; i < XSSZ; i += 32) xs[i] = 0.f;
    const float* xb = x + (size_t)b * 784;
    for (int i = lane; i < 784; i += 32) {
        int r = i / 28, c = i % 28;
        xs[(r + 1) * 32 + (c + 1)] = xb[i];
    }

    // ---- conv1 + relu + maxpool2 into halo-padded h1p ----
    for (int i = lane; i < H1SZ; i += 32) h1p[i] = 0.f;
    for (int it = 0; it < 49; ++it) {              // 49*32 == 1568 outputs exactly
        const int o  = lane + it * 32;
        const int oc = o / 196;
        const int r  = o % 196;
        const int oy = r / 14, ox = r % 14;
        float patch[4][4];
#pragma unroll
        for (int dy = 0; dy < 4; ++dy)
#pragma unroll
            for (int dx = 0; dx < 4; ++dx)
                patch[dy][dx] = xs[(2 * oy + dy) * 32 + (2 * ox + dx)];
        const float* wk = wsh + oc * 9;
        const float bias = wsh[72 + oc];
        float m = -1e30f;
#pragma unroll
        for (int py = 0; py < 2; ++py)
#pragma unroll
            for (int px = 0; px < 2; ++px) {
                float s = bias;
#pragma unroll
                for (int ky = 0; ky < 3; ++ky)
#pragma unroll
                    for (int kx = 0; kx < 3; ++kx)
                        s = fmaf(wk[ky * 3 + kx], patch[py + ky][px + kx], s);
                m = fmaxf(m, s);
            }
        // relu(max) == max(relu); store into halo-padded layout
        h1p[oc * 240 + (oy + 1) * 16 + (ox + 1)] = fmaxf(m, 0.f);
    }
    __syncthreads();

    // ---- conv2 as implicit GEMM with V_WMMA_F32_16X16X4_F32 ----
    // K = c*9 + dy*3 + dx (72 total, 18 steps of 4).
    // A fragment (16x4 f32): lane L -> row M = L&15, Khalf = L>>4,
    //   vgpr0 = K=2h, vgpr1 = K=2h+1.  B fragment mirrored with N = L&15.
    const int khalf = lane >> 4;
    const int n     = lane & 15;

    float bf0[18], bf1[18];
    int   ao0[18], ao1[18];
#pragma unroll
    for (int s = 0; s < 18; ++s) {
        const int k0 = 4 * s + 2 * khalf, k1 = k0 + 1;
        bf0[s] = w2g[n * 72 + k0];               // conv2_w[n][c][dy][dx], flat k
        bf1[s] = w2g[n * 72 + k1];
        const int c0 = k0 / 9, t0 = k0 % 9;
        const int c1 = k1 / 9, t1 = k1 % 9;
        ao0[s] = c0 * 240 + (t0 / 3) * 16 + (t0 % 3);   // halo makes this unpredicated
        ao1[s] = c1 * 240 + (t1 / 3) * 16 + (t1 % 3);
    }
    const float cbias = wsh[80 + n];

    for (int tile = 0; tile < 9; ++tile) {       // 9 tiles of 16 pixels = 144
        const int p  = tile * 16 + n;            // A row m == lane&15
        const int py = p / 12, px = p % 12;
        const int pb = py * 16 + px;             // pixel part of LDS address
        f32x8 acc = {cbias, cbias, cbias, cbias, cbias, cbias, cbias, cbias};
#pragma unroll
        for (int s = 0; s < 18; ++s) {
            f32x2 a, bv;
            a.x  = h1p[ao0[s] + pb];
            a.y  = h1p[ao1[s] + pb];
            bv.x = bf0[s];
            bv.y = bf1[s];
            acc = __builtin_amdgcn_wmma_f32_16x16x4_f32(
                false, a, false, bv, (short)0, acc, false, false);
        }
        // D layout: vgpr r -> M = r + 8*(lane>>4), N = lane&15
#pragma unroll
        for (int r = 0; r < 8; ++r) {
            const int m = r + 8 * khalf;
            c2s[(tile * 16 + m) * 16 + n] = fmaxf(acc[r], 0.f);
        }
    }
    __syncthreads();

    // ---- maxpool2 over 12x12 then mean over 6x6 ----
    if (lane < 16) {
        float s = 0.f;
#pragma unroll
        for (int wy = 0; wy < 6; ++wy)
#pragma unroll
            for (int wx = 0; wx < 6; ++wx) {
                const int p00 = (2 * wy) * 12 + 2 * wx;
                const float a = fmaxf(fmaxf(c2s[p00 * 16 + lane], c2s[(p00 + 1) * 16 + lane]),
                                      fmaxf(c2s[(p00 + 12) * 16 + lane], c2s[(p00 + 13) * 16 + lane]));
                s += a;
            }
        pooled[(size_t)b * 16 + lane] = s * (1.0f / 36.0f);
    }
}

// ---------------------------------------------------------------------------
// Kernel 2: 4-qubit statevector simulation, one thread per batch element.
// Wire w lives on bit (3-w) of the amplitude index. Fully template-unrolled
// so the 16 complex amps stay in VGPRs.
// ---------------------------------------------------------------------------
template<int BIT>
__device__ __forceinline__ void g_apply(float ar[16], float ai[16],
    float u00r, float u00i, float u01r, float u01i,
    float u10r, float u10i, float u11r, float u11i)
{
#pragma unroll
    for (int i0 = 0; i0 < 16; ++i0) {
        if ((i0 & (1 << BIT)) == 0) {
            const int i1 = i0 | (1 << BIT);
            const float a0r = ar[i0], a0i = ai[i0];
            const float a1r = ar[i1], a1i = ai[i1];
            ar[i0] = u00r * a0r - u00i * a0i + u01r * a1r - u01i * a1i;
            ai[i0] = u00r * a0i + u00i * a0r + u01r * a1i + u01i * a1r;
            ar[i1] = u10r * a0r - u10i * a0i + u11r * a1r - u11i * a1i;
            ai[i1] = u10r * a0i + u10i * a0r + u11r * a1i + u11i * a1r;
        }
    }
}
template<int BIT> __device__ __forceinline__ void g_rx(float ar[16], float ai[16], float t) {
    float s, c; __sincosf(0.5f * t, &s, &c);
    g_apply<BIT>(ar, ai, c, 0.f, 0.f, -s, 0.f, -s, c, 0.f);
}
template<int BIT> __device__ __forceinline__ void g_ry(float ar[16], float ai[16], float t) {
    float s, c; __sincosf(0.5f * t, &s, &c);
    g_apply<BIT>(ar, ai, c, 0.f, -s, 0.f, s, 0.f, c, 0.f);
}
template<int BIT> __device__ __forceinline__ void g_rz(float ar[16], float ai[16], float t) {
    float s, c; __sincosf(0.5f * t, &s, &c);
    g_apply<BIT>(ar, ai, c, -s, 0.f, 0.f, 0.f, 0.f, c, s);
}
template<int CBIT, int TBIT>
__device__ __forceinline__ void g_crx(float ar[16], float ai[16], float t) {
    float s, c; __sincosf(0.5f * t, &s, &c);
#pragma unroll
    for (int i0 = 0; i0 < 16; ++i0) {
        if ((i0 & (1 << CBIT)) != 0 && (i0 & (1 << TBIT)) == 0) {
            const int i1 = i0 | (1 << TBIT);
            const float a0r = ar[i0], a0i = ai[i0];
            const float a1r = ar[i1], a1i = ai[i1];
            ar[i0] = c * a0r + s * a1i;  ai[i0] = c * a0i - s * a1r;
            ar[i1] = c * a1r + s * a0i;  ai[i1] = c * a1i - s * a0r;
        }
    }
}

__global__ __launch_bounds__(256)
void quantum_kernel(const float* __restrict__ pooled,
                    const float* __restrict__ rp, const float* __restrict__ qp,
                    float* __restrict__ qout, int B)
{
    const int b = blockIdx.x * blockDim.x + threadIdx.x;
    if (b >= B) return;
    float ar[16], ai[16];
#pragma unroll
    for (int i = 0; i < 16; ++i) { ar[i] = 0.f; ai[i] = 0.f; }
    ar[0] = 1.f;
    const float* pb = pooled + (size_t)b * 16;

    // encoder '4x4_ryzxy': wires 0..3 -> bits 3..0
    g_ry<3>(ar, ai, pb[0]);  g_ry<2>(ar, ai, pb[1]);  g_ry<1>(ar, ai, pb[2]);  g_ry<0>(ar, ai, pb[3]);
    g_rz<3>(ar, ai, pb[4]);  g_rz<2>(ar, ai, pb[5]);  g_rz<1>(ar, ai, pb[6]);  g_rz<0>(ar, ai, pb[7]);
    g_rx<3>(ar, ai, pb[8]);  g_rx<2>(ar, ai, pb[9]);  g_rx<1>(ar, ai, pb[10]); g_rx<0>(ar, ai, pb[11]);
    g_ry<3>(ar, ai, pb[12]); g_ry<2>(ar, ai, pb[13]); g_ry<1>(ar, ai, pb[14]); g_ry<0>(ar, ai, pb[15]);

    // RandomLayer surrogate: rot = i%3 (rx,ry,rz), wire = i%4 -> bit 3-(i%4)
    g_rx<3>(ar, ai, rp[0]);  g_ry<2>(ar, ai, rp[1]);  g_rz<1>(ar, ai, rp[2]);  g_rx<0>(ar, ai, rp[3]);
    g_ry<3>(ar, ai, rp[4]);  g_rz<2>(ar, ai, rp[5]);  g_rx<1>(ar, ai, rp[6]);  g_ry<0>(ar, ai, rp[7]);
    g_rz<3>(ar, ai, rp[8]);  g_rx<2>(ar, ai, rp[9]);  g_ry<1>(ar, ai, rp[10]); g_rz<0>(ar, ai, rp[11]);
    g_rx<3>(ar, ai, rp[12]); g_ry<2>(ar, ai, rp[13]); g_rz<1>(ar, ai, rp[14]); g_rx<0>(ar, ai, rp[15]);
    g_ry<3>(ar, ai, rp[16]); g_rz<2>(ar, ai, rp[17]); g_rx<1>(ar, ai, rp[18]); g_ry<0>(ar, ai, rp[19]);
    g_rz<3>(ar, ai, rp[20]); g_rx<2>(ar, ai, rp[21]); g_ry<1>(ar, ai, rp[22]); g_rz<0>(ar, ai, rp[23]);
    g_rx<3>(ar, ai, rp[24]); g_ry<2>(ar, ai, rp[25]); g_rz<1>(ar, ai, rp[26]); g_rx<0>(ar, ai, rp[27]);
    g_ry<3>(ar, ai, rp[28]); g_rz<2>(ar, ai, rp[29]);

    // trainable layer
    g_rx<3>(ar, ai, qp[0]);          // rx wire0
    g_ry<2>(ar, ai, qp[1]);          // ry wire1
    g_rz<0>(ar, ai, qp[2]);          // rz wire3
    g_crx<3, 1>(ar, ai, qp[3]);      // crx control wire0(bit3) target wire2(bit1)
    // CZ wires (1,3) -> bits (2,0)
#pragma unroll
    for (int i = 0; i < 16; ++i)
        if ((i & 4) && (i & 1)) { ar[i] = -ar[i]; ai[i] = -ai[i]; }
    // H wire3 -> bit0
    { const float r = 0.70710678118654752f;
      g_apply<0>(ar, ai, r, 0.f, r, 0.f, r, 0.f, -r, 0.f); }
    // SX wire2 -> bit1
    g_apply<1>(ar, ai, 0.5f, 0.5f, 0.5f, -0.5f, 0.5f, -0.5f, 0.5f, 0.5f);
    // CNOT control wire3(bit0), target wire0(bit3)
#pragma unroll
    for (int i = 0; i < 16; ++i)
        if ((i & 1) && !(i & 8)) {
            const float tr = ar[i], ti = ai[i];
            ar[i] = ar[i | 8]; ai[i] = ai[i | 8];
            ar[i | 8] = tr;    ai[i | 8] = ti;
        }

    float p[16];
#pragma unroll
    for (int i = 0; i < 16; ++i) p[i] = ar[i] * ar[i] + ai[i] * ai[i];
#pragma unroll
    for (int w = 0; w < 4; ++w) {
        const int bit = 3 - w;
        float e = 0.f;
#pragma unroll
        for (int i = 0; i < 16; ++i) e += (i & (1 << bit)) ? -p[i] : p[i];
        qout[(size_t)b * 4 + w] = e;
    }
}

// ---------------------------------------------------------------------------
// Kernels 3/4: deterministic batch mean/var reduction for BatchNorm1d
// ---------------------------------------------------------------------------
__global__ __launch_bounds__(256)
void bn_reduce1_kernel(const float* __restrict__ q, float* __restrict__ partials, int B)
{
    __shared__ float sd[256 * 8];
    const int t = threadIdx.x;
    float s[4] = {0.f, 0.f, 0.f, 0.f}, ss[4] = {0.f, 0.f, 0.f, 0.f};
    for (int b = blockIdx.x * blockDim.x + t; b < B; b += gridDim.x * blockDim.x) {
#pragma unroll
        for (int k = 0; k < 4; ++k) {
            const float v = q[(size_t)b * 4 + k];
            s[k] += v; ss[k] += v * v;
        }
    }
#pragma unroll
    for (int k = 0; k < 4; ++k) { sd[t * 8 + k] = s[k]; sd[t * 8 + 4 + k] = ss[k]; }
    __syncthreads();
    for (int off = 128; off > 0; off >>= 1) {
        if (t < off)
#pragma unroll
            for (int j = 0; j < 8; ++j) sd[t * 8 + j] += sd[(t + off) * 8 + j];
        __syncthreads();
    }
    if (t == 0)
#pragma unroll
        for (int j = 0; j < 8; ++j) partials[blockIdx.x * 8 + j] = sd[j];
}

__global__ __launch_bounds__(64)
void bn_reduce2_kernel(const float* __restrict__ partials, float* __restrict__ bnstats,
                       int nblk, int B)
{
    const int t = threadIdx.x;
    if (t < 4) {
        float s = 0.f, ss = 0.f;
        for (int i = 0; i < nblk; ++i) { s += partials[i * 8 + t]; ss += partials[i * 8 + 4 + t]; }
        const float mean = s / (float)B;
        const float var  = ss / (float)B - mean * mean;   // biased var (jnp.var)
        bnstats[t]     = mean;
        bnstats[4 + t] = rsqrtf(var + 1e-5f);
    }
}

// ---------------------------------------------------------------------------
// Kernel 5: BN apply + MLP heads (regression + softmax classifier)
// ---------------------------------------------------------------------------
__global__ __launch_bounds__(256)
void head_kernel(const float* __restrict__ q, const float* __restrict__ bnstats,
                 const float* __restrict__ gamma, const float* __restrict__ beta,
                 const float* __restrict__ ew1, const float* __restrict__ eb1,
                 const float* __restrict__ ew2, const float* __restrict__ eb2,
                 const float* __restrict__ ew3, const float* __restrict__ eb3,
                 const float* __restrict__ sw1, const float* __restrict__ sb1,
                 const float* __restrict__ sw2, const float* __restrict__ sb2,
                 float* __restrict__ out, int B)
{
    const int b = blockIdx.x * blockDim.x + threadIdx.x;
    if (b >= B) return;
    float qn[4];
#pragma unroll
    for (int k = 0; k < 4; ++k)
        qn[k] = (q[(size_t)b * 4 + k] - bnstats[k]) * bnstats[4 + k] * gamma[k] + beta[k];

    float e1[8];
#pragma unroll
    for (int i = 0; i < 8; ++i) {
        float t = eb1[i];
#pragma unroll
        for (int k = 0; k < 4; ++k) t = fmaf(ew1[i * 4 + k], qn[k], t);
        e1[i] = tanhf(t);
    }
    float e2[4];
#pragma unroll
    for (int i = 0; i < 4; ++i) {
        float t = eb2[i];
#pragma unroll
        for (int k = 0; k < 8; ++k) t = fmaf(ew2[i * 8 + k], e1[k], t);
        e2[i] = tanhf(t);
    }
    float reg = eb3[0];
#pragma unroll
    for (int k = 0; k < 4; ++k) reg = fmaf(ew3[k], e2[k], reg);

    float s1[4];
#pragma unroll
    for (int i = 0; i < 4; ++i) {
        float t = sb1[i];
#pragma unroll
        for (int k = 0; k < 4; ++k) t = fmaf(sw1[i * 4 + k], qn[k], t);
        s1[i] = tanhf(t);
    }
    float l0 = sb2[0], l1 = sb2[1];
#pragma unroll
    for (int k = 0; k < 4; ++k) { l0 = fmaf(sw2[k], s1[k], l0); l1 = fmaf(sw2[4 + k], s1[k], l1); }
    const float mx = fmaxf(l0, l1);
    const float x0 = __expf(l0 - mx), x1 = __expf(l1 - mx);
    const float inv = 1.f / (x0 + x1);

    out[b] = reg;                        // reg [B,1] first
    out[(size_t)B + 2 * b + 0] = x0 * inv;  // then cls [B,2]
    out[(size_t)B + 2 * b + 1] = x1 * inv;
}

// ---------------------------------------------------------------------------
extern "C" void kernel_launch(void* const* d_in, const int* in_sizes, int n_in,
                              void* d_out, int out_size, void* d_ws, size_t ws_size,
                              hipStream_t stream)
{
    const float* x   = (const float*)d_in[0];
    const float* w1  = (const float*)d_in[1];
    const float* b1  = (const float*)d_in[2];
    const float* w2  = (const float*)d_in[3];
    const float* b2  = (const float*)d_in[4];
    const float* rp  = (const float*)d_in[5];
    const float* qp  = (const float*)d_in[6];
    const float* gam = (const float*)d_in[7];
    const float* bet = (const float*)d_in[8];
    const float* ew1 = (const float*)d_in[9];
    const float* eb1 = (const float*)d_in[10];
    const float* ew2 = (const float*)d_in[11];
    const float* eb2 = (const float*)d_in[12];
    const float* ew3 = (const float*)d_in[13];
    const float* eb3 = (const float*)d_in[14];
    const float* sw1 = (const float*)d_in[15];
    const float* sb1 = (const float*)d_in[16];
    const float* sw2 = (const float*)d_in[17];
    const float* sb2 = (const float*)d_in[18];
    float* out = (float*)d_out;

    const int B = in_sizes[0] / 784;     // 16384

    float* pooled   = (float*)d_ws;              // B*16
    float* q        = pooled + (size_t)B * 16;   // B*4
    float* partials = q + (size_t)B * 4;         // 64*8
    float* bnstats  = partials + 64 * 8;         // 8

    fused_cnn_kernel<<<B / WPB, 64, 0, stream>>>(x, w1, b1, w2, b2, pooled, B);
    quantum_kernel<<<(B + 255) / 256, 256, 0, stream>>>(pooled, rp, qp, q, B);
    bn_reduce1_kernel<<<64, 256, 0, stream>>>(q, partials, B);
    bn_reduce2_kernel<<<1, 64, 0, stream>>>(partials, bnstats, 64, B);
    head_kernel<<<(B + 255) / 256, 256, 0, stream>>>(q, bnstats, gam, bet,
        ew1, eb1, ew2, eb2, ew3, eb3, sw1, sb1, sw2, sb2, out, B);
}